// HGNN_conv4_78099685311015
// MI455X (gfx1250) — compile-verified
//
#include <hip/hip_runtime.h>
#include <stdint.h>

// ---------------- problem constants (match reference) ----------------
#define N_ITEMS   100000
#define N_BASKETS 50000
#define NNZ_CNT   1000000
#define DIM       128
#define NUM_LAYER 3

// ---------------- spmm tiling ----------------
#define LANES            32   // wave32
#define WAVES_PER_BLOCK  8    // 256 threads / block
#define PER_WAVE         16   // nnz handled per wave (NNZ % 16 == 0)

// =====================================================================
// Elementwise helpers (float4, streaming, trivially bandwidth-bound)
// =====================================================================
__global__ __launch_bounds__(256) void copy2_kernel(const float4* __restrict__ s,
                                                    float4* __restrict__ a,
                                                    float4* __restrict__ b, int n4) {
  int i = blockIdx.x * blockDim.x + threadIdx.x;
  if (i < n4) { float4 v = s[i]; a[i] = v; b[i] = v; }
}

__global__ __launch_bounds__(256) void zero_kernel(float4* __restrict__ p, int n4) {
  int i = blockIdx.x * blockDim.x + threadIdx.x;
  if (i < n4) p[i] = make_float4(0.f, 0.f, 0.f, 0.f);
}

__global__ __launch_bounds__(256) void axpy_kernel(float4* __restrict__ acc,
                                                   const float4* __restrict__ add, int n4) {
  int i = blockIdx.x * blockDim.x + threadIdx.x;
  if (i < n4) {
    float4 x = acc[i]; float4 y = add[i];
    x.x += y.x; x.y += y.y; x.z += y.z; x.w += y.w;
    acc[i] = x;
  }
}

__global__ __launch_bounds__(256) void scale_kernel(float4* __restrict__ out,
                                                    int nI4, int nT4) {
  int i = blockIdx.x * blockDim.x + threadIdx.x;
  if (i >= nT4) return;
  float s = (i < nI4) ? 0.25f : (1.0f / 3.0f);
  float4 v = out[i];
  v.x *= s; v.y *= s; v.z *= s; v.w *= s;
  out[i] = v;
}

// =====================================================================
// COO SpMM scatter:  dst[rows[k]] += vals[k] * src[cols[k]]   (rows of DIM)
//
// One wave per 16 nonzeros.
//  - Gather: 16 async 512B row loads to LDS (ASYNCcnt-tracked, in-order),
//    SADDR = scalar row base, VADDR = constant lane*16 offset.
//  - Broadcasts via v_readlane (uniform scalars -> SALU addressing).
//  - Scatter: no-return global_atomic_add_f32 in saddr form into the
//    L2-resident destination matrix.
// =====================================================================
__global__ __launch_bounds__(WAVES_PER_BLOCK * LANES)
void spmm_scatter_kernel(const int* __restrict__ rows,
                         const int* __restrict__ cols,
                         const float* __restrict__ vals,
                         const float* __restrict__ src,
                         float* __restrict__ dst,
                         int nnz) {
  // 8 waves * 16 slots * 512B = 64 KB of LDS per block
  __shared__ float4 tile[WAVES_PER_BLOCK * PER_WAVE * LANES];

  const int lane = threadIdx.x & (LANES - 1);
  const int wave = threadIdx.x >> 5;
  const long long base =
      ((long long)blockIdx.x * WAVES_PER_BLOCK + wave) * PER_WAVE;
  if (base >= nnz) return;  // whole-wave guard; active waves always do 16 nnz

  // Lanes 0..15 preload this wave's 16 (row, col, val) triples.
  int my_row = 0, my_col = 0, my_val_i = 0;
  if (lane < PER_WAVE) {
    my_row   = rows[base + lane];
    my_col   = cols[base + lane];
    my_val_i = __float_as_int(vals[base + lane]);
  }

  // Per-lane constant byte offset within a 512B row (VADDR for async loads).
  const uint32_t voff = (uint32_t)lane * 16u;

  // ---- issue 16 async row-gathers (each lane moves 16B of a 512B row) ----
  #pragma unroll
  for (int k = 0; k < PER_WAVE; ++k) {
    int c = __builtin_amdgcn_readlane(my_col, k);           // uniform scalar
    const float* srow = src + (size_t)(unsigned)c * DIM;    // SALU base
    uint32_t laddr = (uint32_t)(uintptr_t)
        (&tile[(wave * PER_WAVE + k) * LANES + lane]);
    // GVS mode: mem_addr = SADDR(row base) + VADDR(lane*16)
    asm volatile("global_load_async_to_lds_b128 %0, %1, %2"
                 :: "v"(laddr), "v"(voff), "s"(srow)
                 : "memory");
  }

  // ---- consume first 8 rows while the last 8 gathers are still in flight ----
  asm volatile("s_wait_asynccnt 0x8" ::: "memory");
  #pragma unroll
  for (int k = 0; k < PER_WAVE / 2; ++k) {
    float4 v = tile[(wave * PER_WAVE + k) * LANES + lane];
    int    r = __builtin_amdgcn_readlane(my_row, k);        // uniform scalar
    float  a = __int_as_float(__builtin_amdgcn_readlane(my_val_i, k));
    float* d = dst + (size_t)(unsigned)r * DIM + lane * 4;  // saddr-form atomics
    unsafeAtomicAdd(d + 0, v.x * a);
    unsafeAtomicAdd(d + 1, v.y * a);
    unsafeAtomicAdd(d + 2, v.z * a);
    unsafeAtomicAdd(d + 3, v.w * a);
  }

  // ---- drain and consume the remaining 8 rows ----
  asm volatile("s_wait_asynccnt 0x0" ::: "memory");
  #pragma unroll
  for (int k = PER_WAVE / 2; k < PER_WAVE; ++k) {
    float4 v = tile[(wave * PER_WAVE + k) * LANES + lane];
    int    r = __builtin_amdgcn_readlane(my_row, k);
    float  a = __int_as_float(__builtin_amdgcn_readlane(my_val_i, k));
    float* d = dst + (size_t)(unsigned)r * DIM + lane * 4;
    unsafeAtomicAdd(d + 0, v.x * a);
    unsafeAtomicAdd(d + 1, v.y * a);
    unsafeAtomicAdd(d + 2, v.z * a);
    unsafeAtomicAdd(d + 3, v.w * a);
  }
}

// =====================================================================
// Host-side orchestration (all on `stream`, graph-capture safe)
// =====================================================================
extern "C" void kernel_launch(void* const* d_in, const int* in_sizes, int n_in,
                              void* d_out, int out_size, void* d_ws, size_t ws_size,
                              hipStream_t stream) {
  (void)in_sizes; (void)n_in; (void)out_size; (void)ws_size;

  const float* x      = (const float*)d_in[0];
  const int*   b_rows = (const int*)  d_in[1];
  const int*   b_cols = (const int*)  d_in[2];
  const float* b_vals = (const float*)d_in[3];
  const int*   i_rows = (const int*)  d_in[4];
  const int*   i_cols = (const int*)  d_in[5];
  const float* i_vals = (const float*)d_in[6];

  // d_out holds the running sums directly: [item_sum | basket_sum]
  float* item_out   = (float*)d_out;
  float* basket_out = item_out + (size_t)N_ITEMS * DIM;

  // workspace: ping buffers for current item rep and per-layer basket rep
  float* cur    = (float*)d_ws;                       // N_ITEMS   * DIM
  float* basket = cur + (size_t)N_ITEMS * DIM;        // N_BASKETS * DIM

  const int nI4 = N_ITEMS   * DIM / 4;  // 3,200,000 float4
  const int nB4 = N_BASKETS * DIM / 4;  // 1,600,000 float4
  const int nT4 = nI4 + nB4;
  const int TB  = 256;

  const int spmm_blocks =
      (NNZ_CNT / PER_WAVE + WAVES_PER_BLOCK - 1) / WAVES_PER_BLOCK;  // 7813

  // item_sum = x (layer-0 term), cur = x, basket_sum = 0
  copy2_kernel<<<(nI4 + TB - 1) / TB, TB, 0, stream>>>(
      (const float4*)x, (float4*)item_out, (float4*)cur, nI4);
  zero_kernel<<<(nB4 + TB - 1) / TB, TB, 0, stream>>>((float4*)basket_out, nB4);

  for (int l = 0; l < NUM_LAYER; ++l) {
    // basket = B @ cur
    zero_kernel<<<(nB4 + TB - 1) / TB, TB, 0, stream>>>((float4*)basket, nB4);
    spmm_scatter_kernel<<<spmm_blocks, TB, 0, stream>>>(
        b_rows, b_cols, b_vals, cur, basket, NNZ_CNT);
    // basket_sum += basket
    axpy_kernel<<<(nB4 + TB - 1) / TB, TB, 0, stream>>>(
        (float4*)basket_out, (const float4*)basket, nB4);
    // cur = I @ basket
    zero_kernel<<<(nI4 + TB - 1) / TB, TB, 0, stream>>>((float4*)cur, nI4);
    spmm_scatter_kernel<<<spmm_blocks, TB, 0, stream>>>(
        i_rows, i_cols, i_vals, basket, cur, NNZ_CNT);
    // item_sum += cur
    axpy_kernel<<<(nI4 + TB - 1) / TB, TB, 0, stream>>>(
        (float4*)item_out, (const float4*)cur, nI4);
  }

  // item /= 4, basket /= 3
  scale_kernel<<<(nT4 + TB - 1) / TB, TB, 0, stream>>>((float4*)d_out, nI4, nT4);
}